// GraphSAGE_86990267613314
// MI455X (gfx1250) — compile-verified
//
#include <hip/hip_runtime.h>
#include <hip/hip_bf16.h>
#include <stdint.h>

#define NN 100000
#define NE 3200000
#define D  128
#define SA_STRIDE 132   // 128 + 4 pad: LDS column reads hit distinct banks

typedef __attribute__((ext_vector_type(2))) float v2f;
typedef __attribute__((ext_vector_type(8))) float v8f;

// ---------------------------------------------------------------- degree
__global__ __launch_bounds__(256) void degree_kernel(
    const long long* __restrict__ dst, float* __restrict__ cnt, int nE) {
  int e = blockIdx.x * 256 + threadIdx.x;
  if (e < nE) atomicAdd(cnt + dst[e], 1.0f);
}

// ---------------------------------------------------------------- scatter-sum
// 32 lanes per edge; each lane moves one float4 (128 floats per row total).
__global__ __launch_bounds__(256) void scatter_kernel(
    const float* __restrict__ X,
    const long long* __restrict__ src,
    const long long* __restrict__ dst,
    float* __restrict__ summed, int nE) {
  int e    = blockIdx.x * 8 + (threadIdx.x >> 5);
  int lane = threadIdx.x & 31;
  if (e >= nE) return;
  long long s = src[e];
  long long d = dst[e];
  const float4* srow = (const float4*)(X + s * (long long)D);
  float4 v = srow[lane];                       // L2-resident gather (51MB < 192MB L2)
  float* drow = summed + d * (long long)D + (long long)lane * 4;
  atomicAdd(drow + 0, v.x);
  atomicAdd(drow + 1, v.y);
  atomicAdd(drow + 2, v.z);
  atomicAdd(drow + 3, v.w);
}

// ---------------------------------------------------------------- fused SAGE GEMM
// out = relu( (summed/max(cnt,1)) @ Wl + X @ Wr + bias )
// Block: 16 output rows, 8 waves x 16-col N-tiles. K=128 -> 32 steps of
// v_wmma_f32_16x16x4_f32, two WMMAs per step (Wl path + Wr path).
// A tiles staged into LDS with GLOBAL_LOAD_ASYNC_TO_LDS_B128 (ASYNCcnt path).
__global__ __launch_bounds__(256) void sage_gemm_kernel(
    const float* __restrict__ X,
    const float* __restrict__ summed,
    const float* __restrict__ cnt,
    const float* __restrict__ Wl,
    const float* __restrict__ Wr,
    const float* __restrict__ bias,
    float* __restrict__ out, int nN) {
  __shared__ float sAl[16 * SA_STRIDE];   // summed neighbors (raw; scaled at use)
  __shared__ float sAr[16 * SA_STRIDE];   // self features
  __shared__ float sInv[16];

  const int tid = threadIdx.x;
  const int m0  = blockIdx.x * 16;

  __builtin_prefetch(Wl, 0, 1);           // -> global_prefetch_b8
  __builtin_prefetch(Wr, 0, 1);

  if (tid < 16) {
    float c = cnt[m0 + tid];
    sInv[tid] = 1.0f / fmaxf(c, 1.0f);
  }

  // Async-stage A tiles: 16 rows x 128 cols, b128 per lane.
  // 512 chunks / matrix, 256 threads -> 2 chunks each per matrix.
  {
    const unsigned long long baseS = (unsigned long long)summed;
    const unsigned long long baseX = (unsigned long long)X;
#pragma unroll
    for (int i = tid; i < 16 * 32; i += 256) {
      int row = i >> 5;
      int c4  = i & 31;
      unsigned goff = (unsigned)(((m0 + row) * D + c4 * 4) * 4);  // byte offset (<52MB)
      unsigned ldsL = (unsigned)(uintptr_t)&sAl[row * SA_STRIDE + c4 * 4];
      unsigned ldsR = (unsigned)(uintptr_t)&sAr[row * SA_STRIDE + c4 * 4];
      asm volatile("global_load_async_to_lds_b128 %0, %1, %2"
                   :: "v"(ldsL), "v"(goff), "s"(baseS) : "memory");
      asm volatile("global_load_async_to_lds_b128 %0, %1, %2"
                   :: "v"(ldsR), "v"(goff), "s"(baseX) : "memory");
    }
  }
  asm volatile("s_wait_asynccnt 0x0" ::: "memory");
  __syncthreads();

  const int wave = tid >> 5;
  const int lane = tid & 31;
  const int n0   = wave * 16;
  const int mrow = lane & 15;          // A: M = lane%16
  const int kb2  = (lane >> 4) * 2;    // A/B: K sub-pair 0 or 2
  const int ncol = n0 + (lane & 15);   // B/C: N = lane%16 within tile

  const float invm = sInv[mrow];       // mean scale for this lane's A row

  v8f acc = {};
#pragma unroll 8
  for (int kk = 0; kk < 32; ++kk) {
    const int k = kk * 4 + kb2;
    v2f al, ar, bl, br;
    al.x = sAl[mrow * SA_STRIDE + k] * invm;
    al.y = sAl[mrow * SA_STRIDE + k + 1] * invm;
    ar.x = sAr[mrow * SA_STRIDE + k];
    ar.y = sAr[mrow * SA_STRIDE + k + 1];
    bl.x = Wl[(long long)k * D + ncol];
    bl.y = Wl[(long long)(k + 1) * D + ncol];
    br.x = Wr[(long long)k * D + ncol];
    br.y = Wr[(long long)(k + 1) * D + ncol];
    acc = __builtin_amdgcn_wmma_f32_16x16x4_f32(
        false, al, false, bl, (short)0, acc, false, false);
    acc = __builtin_amdgcn_wmma_f32_16x16x4_f32(
        false, ar, false, br, (short)0, acc, false, false);
  }

  // C/D layout: VGPR r -> M = r (lanes 0-15), M = r+8 (lanes 16-31), N = lane%16.
  const float bv    = bias[ncol];
  const int   rbase = (lane >> 4) * 8;
#pragma unroll
  for (int r = 0; r < 8; ++r) {
    float v = acc[r] + bv;
    v = fmaxf(v, 0.0f);
    out[(long long)(m0 + rbase + r) * D + ncol] = v;
  }
}

// ---------------------------------------------------------------- launch
extern "C" void kernel_launch(void* const* d_in, const int* in_sizes, int n_in,
                              void* d_out, int out_size, void* d_ws, size_t ws_size,
                              hipStream_t stream) {
  const float*     node_fts = (const float*)d_in[0];
  const long long* ei       = (const long long*)d_in[1];   // int64 edge_index [2, E]
  const float*     Wl1      = (const float*)d_in[2];
  const float*     Wr1      = (const float*)d_in[3];
  const float*     b1       = (const float*)d_in[4];
  const float*     Wl2      = (const float*)d_in[5];
  const float*     Wr2      = (const float*)d_in[6];
  const float*     b2       = (const float*)d_in[7];
  float*           out      = (float*)d_out;

  const long long* src = ei;
  const long long* dst = ei + NE;

  float* summed = (float*)d_ws;                      // [NN, D]
  float* h1     = summed + (size_t)NN * D;           // [NN, D]
  float* cnt    = h1 + (size_t)NN * D;               // [NN]

  // Degrees (same for both layers).
  hipMemsetAsync(cnt, 0, (size_t)NN * sizeof(float), stream);
  degree_kernel<<<(NE + 255) / 256, 256, 0, stream>>>(dst, cnt, NE);

  // Layer 1: node_fts -> h1
  hipMemsetAsync(summed, 0, (size_t)NN * D * sizeof(float), stream);
  scatter_kernel<<<NE / 8, 256, 0, stream>>>(node_fts, src, dst, summed, NE);
  sage_gemm_kernel<<<NN / 16, 256, 0, stream>>>(node_fts, summed, cnt,
                                                Wl1, Wr1, b1, h1, NN);

  // Layer 2: h1 -> out
  hipMemsetAsync(summed, 0, (size_t)NN * D * sizeof(float), stream);
  scatter_kernel<<<NE / 8, 256, 0, stream>>>(h1, src, dst, summed, NE);
  sage_gemm_kernel<<<NN / 16, 256, 0, stream>>>(h1, summed, cnt,
                                                Wl2, Wr2, b2, out, NN);
}